// EM_11373073400042
// MI455X (gfx1250) — compile-verified
//
#include <hip/hip_runtime.h>
#include <hip/hip_bf16.h>

// Problem constants (from reference setup_inputs)
#define BB 16
#define CC 512
#define NN 4096   // H*W = 64*64
#define KK 64
#define EPSF 1e-6f

typedef __attribute__((ext_vector_type(2))) float v2f;
typedef __attribute__((ext_vector_type(8))) float v8f;
typedef int v4i __attribute__((vector_size(4 * sizeof(int))));

// ---------------------------------------------------------------------------
// CDNA5 async global->LDS copy (ASYNCcnt-tracked, bypasses VGPR staging).
// Param 0 is a global-AS v4i*, param 1 the LDS-AS v4i* (per compiler probe).
// Guarded: falls back to synchronous staging if the builtin is absent.
// ---------------------------------------------------------------------------
#if defined(__has_builtin) && __has_builtin(__builtin_amdgcn_global_load_async_to_lds_b128)
#define ASYNC_LDS 1
#else
#define ASYNC_LDS 0
#endif

__device__ __forceinline__ void cp16(void* lds_dst, const void* gsrc) {
#if ASYNC_LDS
    __builtin_amdgcn_global_load_async_to_lds_b128(
        (__attribute__((address_space(1))) v4i*)(void*)gsrc,
        (__attribute__((address_space(3))) v4i*)lds_dst,
        /*offset=*/0, /*cpol=*/0);
#else
    *(float4*)lds_dst = *(const float4*)gsrc;
#endif
}

__device__ __forceinline__ void wait_async() {
#if ASYNC_LDS
#if __has_builtin(__builtin_amdgcn_s_wait_asynccnt)
    __builtin_amdgcn_s_wait_asynccnt(0);
#else
    asm volatile("s_wait_asynccnt 0x0" ::: "memory");
#endif
#endif
}

// D = A(16x4 f32) * B(4x16 f32) + C(16x16 f32)   -- full fp32 matrix pipe
__device__ __forceinline__ v8f wmma4(v2f a, v2f b, v8f c) {
    return __builtin_amdgcn_wmma_f32_16x16x4_f32(
        /*neg_a=*/false, a, /*neg_b=*/false, b,
        /*c_mod=*/(short)0, c, /*reuse_a=*/false, /*reuse_b=*/false);
}

// ---------------------------------------------------------------------------
// Kernel 1: 1x1 conv == per-batch GEMM  xt[o,n] = sum_c w[o,c]*x[c,n] + b[o]
// 128x128 tile / WG (8 waves), K-chunks of 32, double-buffered async LDS.
// ---------------------------------------------------------------------------
__global__ __launch_bounds__(256) void conv_gemm(
    const float* __restrict__ x, const float* __restrict__ w,
    const float* __restrict__ bias, float* __restrict__ xt)
{
    const int b      = blockIdx.z;
    const int o_tile = blockIdx.y * 128;
    const int n_tile = blockIdx.x * 128;
    const float* xb  = x  + (size_t)b * CC * NN;
    float*       xtb = xt + (size_t)b * CC * NN;

    __shared__ float As[2][128][36];   // w tile [o][c], pad->36
    __shared__ float Bs[2][32][132];   // x tile [c][n], pad->132

    const int t    = threadIdx.x;
    const int lane = t & 31;
    const int wv   = t >> 5;
    const int m0w  = (wv >> 2) * 64;  // wave row offset (o)
    const int n0w  = (wv & 3) * 32;   // wave col offset (n)

    v8f acc[4][2];
    #pragma unroll
    for (int i = 0; i < 4; i++)
        #pragma unroll
        for (int j = 0; j < 2; j++) acc[i][j] = {};

    auto issue = [&](int pb, int c0) {
        #pragma unroll
        for (int i = 0; i < 4; i++) {           // 128x32 floats of W
            int idx = t + i * 256;
            int row = idx >> 3, col4 = (idx & 7) * 4;
            cp16(&As[pb][row][col4], w + (size_t)(o_tile + row) * CC + c0 + col4);
        }
        #pragma unroll
        for (int i = 0; i < 4; i++) {           // 32x128 floats of X
            int idx = t + i * 256;
            int row = idx >> 5, col4 = (idx & 31) * 4;
            cp16(&Bs[pb][row][col4], xb + (size_t)(c0 + row) * NN + n_tile + col4);
        }
    };

    issue(0, 0);
    int pb = 0;
    for (int c0 = 0; c0 < CC; c0 += 32) {
        wait_async();
        __syncthreads();
        if (c0 + 32 < CC) issue(pb ^ 1, c0 + 32);
        #pragma unroll
        for (int kk = 0; kk < 32; kk += 4) {
            const int krow = kk + ((lane >> 4) << 1);  // lanes0-15: K=kk,kk+1; 16-31: kk+2,kk+3
            v2f a[4], bf[2];
            #pragma unroll
            for (int i = 0; i < 4; i++)
                a[i] = *(const v2f*)&As[pb][m0w + i * 16 + (lane & 15)][krow];
            #pragma unroll
            for (int j = 0; j < 2; j++) {
                int col = n0w + j * 16 + (lane & 15);
                bf[j].x = Bs[pb][krow][col];
                bf[j].y = Bs[pb][krow + 1][col];
            }
            #pragma unroll
            for (int i = 0; i < 4; i++)
                #pragma unroll
                for (int j = 0; j < 2; j++)
                    acc[i][j] = wmma4(a[i], bf[j], acc[i][j]);
        }
        pb ^= 1;
    }

    const int colb   = lane & 15;
    const int rowoff = (lane >> 4) * 8;
    #pragma unroll
    for (int i = 0; i < 4; i++)
        #pragma unroll
        for (int j = 0; j < 2; j++) {
            int ncol = n_tile + n0w + j * 16 + colb;
            #pragma unroll
            for (int r = 0; r < 8; r++) {
                int orow = o_tile + m0w + i * 16 + rowoff + r;
                xtb[(size_t)orow * NN + ncol] = acc[i][j][r] + bias[orow];
            }
        }
}

// ---------------------------------------------------------------------------
// Kernel 2: z[n,k] = sum_c xt[c,n]*mu[c,k]; softmax over k; colsum atomics;
// store softmaxed z. Tile 128(n) x 64(k=all) -> softmax fully in-WG.
// ---------------------------------------------------------------------------
__global__ __launch_bounds__(256) void z_softmax(
    const float* __restrict__ xt, const float* __restrict__ mu,
    float* __restrict__ zbuf, float* __restrict__ colsum)
{
    const int b      = blockIdx.y;
    const int n_tile = blockIdx.x * 128;
    const float* xtb = xt + (size_t)b * CC * NN;
    const float* mub = mu + (size_t)b * CC * KK;

    __shared__ float As[2][32][132];  // xt chunk [c][n]
    __shared__ float Bs[2][32][68];   // mu chunk [c][k]
    __shared__ float Zs[128][68];     // z tile  [n][k]

    const int t = threadIdx.x, lane = t & 31, wv = t >> 5;
    const int m0w = (wv >> 1) * 32;  // n offset
    const int k0w = (wv & 1) * 32;   // k offset

    v8f acc[2][2];
    #pragma unroll
    for (int i = 0; i < 2; i++)
        #pragma unroll
        for (int j = 0; j < 2; j++) acc[i][j] = {};

    auto issue = [&](int pb, int c0) {
        #pragma unroll
        for (int i = 0; i < 4; i++) {           // 32x128 xt
            int idx = t + i * 256;
            int row = idx >> 5, col4 = (idx & 31) * 4;
            cp16(&As[pb][row][col4], xtb + (size_t)(c0 + row) * NN + n_tile + col4);
        }
        #pragma unroll
        for (int i = 0; i < 2; i++) {           // 32x64 mu
            int idx = t + i * 256;
            int row = idx >> 4, col4 = (idx & 15) * 4;
            cp16(&Bs[pb][row][col4], mub + (size_t)(c0 + row) * KK + col4);
        }
    };

    issue(0, 0);
    int pb = 0;
    for (int c0 = 0; c0 < CC; c0 += 32) {
        wait_async();
        __syncthreads();
        if (c0 + 32 < CC) issue(pb ^ 1, c0 + 32);
        #pragma unroll
        for (int kk = 0; kk < 32; kk += 4) {
            const int krow = kk + ((lane >> 4) << 1);
            v2f a[2], bf[2];
            #pragma unroll
            for (int i = 0; i < 2; i++) {       // A[m=n][kdim=c] read from [c][n] layout
                int col = m0w + i * 16 + (lane & 15);
                a[i].x = As[pb][krow][col];
                a[i].y = As[pb][krow + 1][col];
            }
            #pragma unroll
            for (int j = 0; j < 2; j++) {
                int col = k0w + j * 16 + (lane & 15);
                bf[j].x = Bs[pb][krow][col];
                bf[j].y = Bs[pb][krow + 1][col];
            }
            #pragma unroll
            for (int i = 0; i < 2; i++)
                #pragma unroll
                for (int j = 0; j < 2; j++)
                    acc[i][j] = wmma4(a[i], bf[j], acc[i][j]);
        }
        pb ^= 1;
    }

    const int colb = lane & 15, rowoff = (lane >> 4) * 8;
    #pragma unroll
    for (int i = 0; i < 2; i++)
        #pragma unroll
        for (int j = 0; j < 2; j++)
            #pragma unroll
            for (int r = 0; r < 8; r++)
                Zs[m0w + i * 16 + rowoff + r][k0w + j * 16 + colb] = acc[i][j][r];
    __syncthreads();

    if (t < 128) {                       // softmax over k, one thread per row
        float mx = -3.0e38f;
        #pragma unroll
        for (int k = 0; k < KK; k++) mx = fmaxf(mx, Zs[t][k]);
        float s = 0.f;
        #pragma unroll
        for (int k = 0; k < KK; k++) { float e = __expf(Zs[t][k] - mx); Zs[t][k] = e; s += e; }
        float inv = 1.f / s;
        #pragma unroll
        for (int k = 0; k < KK; k++) Zs[t][k] *= inv;
    }
    __syncthreads();

    if (t < KK) {                        // partial column sums -> global atomics
        float s = 0.f;
        #pragma unroll 8
        for (int r = 0; r < 128; r++) s += Zs[r][t];
        atomicAdd(&colsum[b * KK + t], s);
    }

    #pragma unroll
    for (int i = 0; i < 8; i++) {        // coalesced store of z tile
        int idx = t + i * 256;
        int row = idx >> 4, col4 = (idx & 15) * 4;
        *(float4*)(zbuf + ((size_t)b * NN + n_tile + row) * KK + col4) =
            *(const float4*)&Zs[row][col4];
    }
}

// ---------------------------------------------------------------------------
// Kernel 3: Y[c,k] = sum_n xt[c,n] * z[n,k].  Tile 128(c) x 64(k), loop N.
// ---------------------------------------------------------------------------
__global__ __launch_bounds__(256) void mu_gemm(
    const float* __restrict__ xt, const float* __restrict__ zbuf,
    float* __restrict__ Y)
{
    const int b      = blockIdx.y;
    const int c_tile = blockIdx.x * 128;
    const float* xtb = xt   + (size_t)b * CC * NN;
    const float* zb  = zbuf + (size_t)b * NN * KK;

    __shared__ float As[2][128][36];   // xt tile [c][n-chunk]
    __shared__ float Bs[2][32][68];    // z tile  [n][k]

    const int t = threadIdx.x, lane = t & 31, wv = t >> 5;
    const int m0w = (wv >> 1) * 32;  // c offset
    const int k0w = (wv & 1) * 32;   // k offset

    v8f acc[2][2];
    #pragma unroll
    for (int i = 0; i < 2; i++)
        #pragma unroll
        for (int j = 0; j < 2; j++) acc[i][j] = {};

    auto issue = [&](int pb, int n0) {
        #pragma unroll
        for (int i = 0; i < 4; i++) {           // 128x32 xt (contig in n)
            int idx = t + i * 256;
            int row = idx >> 3, col4 = (idx & 7) * 4;
            cp16(&As[pb][row][col4], xtb + (size_t)(c_tile + row) * NN + n0 + col4);
        }
        #pragma unroll
        for (int i = 0; i < 2; i++) {           // 32x64 z (contig in k)
            int idx = t + i * 256;
            int row = idx >> 4, col4 = (idx & 15) * 4;
            cp16(&Bs[pb][row][col4], zb + (size_t)(n0 + row) * KK + col4);
        }
    };

    issue(0, 0);
    int pb = 0;
    for (int n0 = 0; n0 < NN; n0 += 32) {
        wait_async();
        __syncthreads();
        if (n0 + 32 < NN) issue(pb ^ 1, n0 + 32);
        #pragma unroll
        for (int kk = 0; kk < 32; kk += 4) {
            const int krow = kk + ((lane >> 4) << 1);
            v2f a[2], bf[2];
            #pragma unroll
            for (int i = 0; i < 2; i++)
                a[i] = *(const v2f*)&As[pb][m0w + i * 16 + (lane & 15)][krow];
            #pragma unroll
            for (int j = 0; j < 2; j++) {
                int col = k0w + j * 16 + (lane & 15);
                bf[j].x = Bs[pb][krow][col];
                bf[j].y = Bs[pb][krow + 1][col];
            }
            #pragma unroll
            for (int i = 0; i < 2; i++)
                #pragma unroll
                for (int j = 0; j < 2; j++)
                    acc[i][j] = wmma4(a[i], bf[j], acc[i][j]);
        }
        pb ^= 1;
    }

    const int colb = lane & 15, rowoff = (lane >> 4) * 8;
    #pragma unroll
    for (int i = 0; i < 2; i++)
        #pragma unroll
        for (int j = 0; j < 2; j++)
            #pragma unroll
            for (int r = 0; r < 8; r++) {
                int c = c_tile + m0w + i * 16 + rowoff + r;
                int k = k0w + j * 16 + colb;
                Y[((size_t)b * CC + c) * KK + k] = acc[i][j][r];
            }
}

// ---------------------------------------------------------------------------
// Kernel 4: mu[c,k] = l2norm_c( Y[c,k] / (eps + colsum[k]) ).
// One WG per (b,k); LDS tree reduction over C=512.
// ---------------------------------------------------------------------------
__global__ __launch_bounds__(256) void mu_norm(
    const float* __restrict__ Y, const float* __restrict__ colsum,
    float* __restrict__ mu_out)
{
    const int b = blockIdx.y, k = blockIdx.x, t = threadIdx.x;
    __shared__ float red[256];

    const float inv = 1.f / (EPSF + colsum[b * KK + k]);
    const size_t base = (size_t)b * CC * KK + k;
    float v0 = Y[base + (size_t)t * KK] * inv;
    float v1 = Y[base + (size_t)(t + 256) * KK] * inv;
    red[t] = v0 * v0 + v1 * v1;
    __syncthreads();
    for (int s = 128; s > 0; s >>= 1) {
        if (t < s) red[t] += red[t + s];
        __syncthreads();
    }
    const float invn = 1.f / (EPSF + sqrtf(red[0]));
    mu_out[base + (size_t)t * KK]         = v0 * invn;
    mu_out[base + (size_t)(t + 256) * KK] = v1 * invn;
}

// ---------------------------------------------------------------------------
// Small utilities
// ---------------------------------------------------------------------------
__global__ void init_mu(const float* __restrict__ mu0, float* __restrict__ mu_cur)
{
    int i = blockIdx.x * blockDim.x + threadIdx.x;   // over B*C*K
    mu_cur[i] = mu0[i % (CC * KK)];
}

__global__ void zero_f(float* __restrict__ p, int n)
{
    int i = blockIdx.x * blockDim.x + threadIdx.x;
    if (i < n) p[i] = 0.f;
}

// ---------------------------------------------------------------------------
extern "C" void kernel_launch(void* const* d_in, const int* in_sizes, int n_in,
                              void* d_out, int out_size, void* d_ws, size_t ws_size,
                              hipStream_t stream)
{
    const float* x      = (const float*)d_in[0];   // [16,512,64,64]
    const float* conv_w = (const float*)d_in[1];   // [512,512]
    const float* conv_b = (const float*)d_in[2];   // [512]
    const float* mu0    = (const float*)d_in[3];   // [1,512,64]

    float* out_mu  = (float*)d_out;                         // [16,512,64]
    float* x_trans = out_mu + (size_t)BB * CC * KK;         // [16,512,4096]

    // workspace layout (~21 MB)
    float* zbuf   = (float*)d_ws;                           // B*N*K
    float* colsum = zbuf + (size_t)BB * NN * KK;            // B*K
    float* Ybuf   = colsum + BB * KK;                       // B*C*K
    float* mu_cur = Ybuf + (size_t)BB * CC * KK;            // B*C*K

    conv_gemm<<<dim3(NN / 128, CC / 128, BB), 256, 0, stream>>>(x, conv_w, conv_b, x_trans);
    init_mu<<<(BB * CC * KK) / 256, 256, 0, stream>>>(mu0, mu_cur);

    for (int s = 0; s < 3; s++) {
        zero_f<<<(BB * KK + 255) / 256, 256, 0, stream>>>(colsum, BB * KK);
        z_softmax<<<dim3(NN / 128, BB), 256, 0, stream>>>(x_trans, mu_cur, zbuf, colsum);
        mu_gemm<<<dim3(CC / 128, BB), 256, 0, stream>>>(x_trans, zbuf, Ybuf);
        float* dst = (s == 2) ? out_mu : mu_cur;
        mu_norm<<<dim3(KK, BB), 256, 0, stream>>>(Ybuf, colsum, dst);
    }
}